// ModelGAT_64244120814044
// MI455X (gfx1250) — compile-verified
//
#include <hip/hip_runtime.h>
#include <hip/hip_bf16.h>
#include <math.h>
#include <stdint.h>

#define NEG_SLOPE 0.2f

typedef __attribute__((ext_vector_type(2))) float v2f;
typedef __attribute__((ext_vector_type(8))) float v8f;

// ==========================================================================
// WMMA GEMM:  C[M x 64] = act(A[M x K] @ W[K x 64] (+ bias))
// fp32 path via V_WMMA_F32_16X16X4_F32 (exact vs fp32 reference).
// Block = 256 thr = 8 waves; block tile = 32 rows x 64 cols;
// wave tile = 16x16, K-loop in steps of 4.
//
// W (tiny: <=32KB) is staged ONCE per block into LDS, transposed with a +2
// pad, using GLOBAL_LOAD_ASYNC_TO_LDS_B32 (per-lane LDS dest VGPR performs
// the transpose scatter; ASYNCcnt-tracked, no VGPR round trip). Inner loop
// then reads the B fragment as one ds_load_b64 with an immediate offset.
//
// A-frag layout (ISA 7.12.2, 32-bit A 16x4): lane l<16 holds row M=l,
// K={k,k+1}; lane l>=16 holds row M=l-16, K={k+2,k+3} -> one v2f load.
// B-frag (4x16) mirrored. C/D v8f: row = rowBase + r + 8*(lane>=16),
// col = colBase + (lane&15).
// ==========================================================================
template<int K, bool RELU, bool BIAS>
__global__ __launch_bounds__(256)
void gemm_wmma_n64(const float* __restrict__ A, const float* __restrict__ W,
                   const float* __restrict__ bias, float* __restrict__ C, int M)
{
  constexpr int KP = K + 2;                 // padded row stride (bank-conflict free)
  __shared__ float wt[64 * KP];             // wt[n*KP + k] = W[k*64 + n]

  const int wave   = threadIdx.x >> 5;
  const int lane   = threadIdx.x & 31;
  const int lhalf  = lane >> 4;             // 0: lanes 0-15, 1: lanes 16-31
  const int l16    = lane & 15;
  const int rowBase = blockIdx.x * 32 + (wave >> 2) * 16;
  const int colBase = (wave & 3) * 16;

  // ---- async-stage W[K x 64] -> LDS, transposed ----
  {
    const unsigned lds_base = (unsigned)(uintptr_t)wt;   // low 32b = LDS offset
#pragma unroll 4
    for (int idx = threadIdx.x; idx < K * 64; idx += 256) {
      const int k = idx >> 6, n = idx & 63;
      const unsigned ldsoff = lds_base + (unsigned)((n * KP + k) * 4);
      const float* g = W + idx;
      asm volatile("global_load_async_to_lds_b32 %0, %1, off"
                   :: "v"(ldsoff), "v"(g) : "memory");
    }
#if __has_builtin(__builtin_amdgcn_s_wait_asynccnt)
    __builtin_amdgcn_s_wait_asynccnt(0);
#else
    asm volatile("s_wait_asynccnt 0x0" ::: "memory");
#endif
  }
  __syncthreads();

  int m = rowBase + l16;
  if (m >= M) m = M - 1;                    // clamp; M%32==0 in practice
  const int n = colBase + l16;

  const float* __restrict__ arow = A + (size_t)m * K;
  const float* __restrict__ wrow = &wt[n * KP];
  v8f acc = {};
#pragma unroll
  for (int k = 0; k < K; k += 4) {
    const int kb = k + 2 * lhalf;
    v2f av = *(const v2f*)(arow + kb);      // 8B aligned (kb even)
    v2f bv = *(const v2f*)(wrow + kb);      // ds_load_b64, imm offset
    acc = __builtin_amdgcn_wmma_f32_16x16x4_f32(
        /*neg_a=*/false, av, /*neg_b=*/false, bv,
        /*c_mod=*/(short)0, acc, /*reuse_a=*/false, /*reuse_b=*/false);
  }

  const float bcol = BIAS ? bias[n] : 0.0f;
#pragma unroll
  for (int r = 0; r < 8; ++r) {
    const int row = rowBase + r + 8 * lhalf;
    float v = acc[r] + bcol;
    if (RELU) v = fmaxf(v, 0.0f);
    if (row < M) C[(size_t)row * 64 + n] = v;
  }
}

// ==========================================================================
// Per-node attention scores: a_s[v] = h[v,:]·att_src ; a_d[v] = h[v,:]·att_dst
// One wave32 per node, 2 features per lane, shfl_xor tree reduction.
// ==========================================================================
__global__ __launch_bounds__(256)
void att_scores(const float* __restrict__ h, const float* __restrict__ att_src,
                const float* __restrict__ att_dst,
                float* __restrict__ a_s, float* __restrict__ a_d, int n_nodes)
{
  const int node = (int)((blockIdx.x * blockDim.x + threadIdx.x) >> 5);
  const int lane = threadIdx.x & 31;
  if (node >= n_nodes) return;
  const float* hr = h + (size_t)node * 64;
  const float h0 = hr[lane], h1 = hr[lane + 32];
  float s = h0 * att_src[lane] + h1 * att_src[lane + 32];
  float d = h0 * att_dst[lane] + h1 * att_dst[lane + 32];
#pragma unroll
  for (int off = 16; off; off >>= 1) {
    s += __shfl_xor(s, off, 32);
    d += __shfl_xor(d, off, 32);
  }
  if (lane == 0) { a_s[node] = s; a_d[node] = d; }
}

// Order-preserving float<->uint encoding for atomicMax-based segment max.
__device__ __forceinline__ unsigned fenc(float f) {
  unsigned u = __float_as_uint(f);
  return (u & 0x80000000u) ? ~u : (u | 0x80000000u);
}
__device__ __forceinline__ float fdec(unsigned u) {
  return __uint_as_float((u & 0x80000000u) ? (u ^ 0x80000000u) : ~u);
}

__device__ __forceinline__ void edge_sd(const int* __restrict__ ei, int E,
                                        int e, int& s, int& d) {
  if (e < E) { s = ei[e]; d = ei[E + e]; }   // edge_index[0][e], [1][e]
  else       { s = d = e - E; }              // appended self loop
}

// Pass 1: segment max of leaky_relu(a_s[src]+a_d[dst]) over dst.
__global__ __launch_bounds__(256)
void edge_max(const int* __restrict__ ei, int E, int n_nodes,
              const float* __restrict__ a_s, const float* __restrict__ a_d,
              unsigned* __restrict__ emax_u)
{
  const int e = blockIdx.x * blockDim.x + threadIdx.x;
  if (e >= E + n_nodes) return;
  int s, d; edge_sd(ei, E, e, s, d);
  float v = a_s[s] + a_d[d];
  v = (v > 0.0f) ? v : NEG_SLOPE * v;
  atomicMax(emax_u + d, fenc(v));
}

// Pass 2: ex[e] = exp(val - max[dst]); denom[dst] += ex[e].
__global__ __launch_bounds__(256)
void edge_expsum(const int* __restrict__ ei, int E, int n_nodes,
                 const float* __restrict__ a_s, const float* __restrict__ a_d,
                 const unsigned* __restrict__ emax_u,
                 float* __restrict__ exb, float* __restrict__ denom)
{
  const int e = blockIdx.x * blockDim.x + threadIdx.x;
  if (e >= E + n_nodes) return;
  int s, d; edge_sd(ei, E, e, s, d);
  float v = a_s[s] + a_d[d];
  v = (v > 0.0f) ? v : NEG_SLOPE * v;
  const float ex = expf(v - fdec(emax_u[d]));
  exb[e] = ex;
  atomicAdd(denom + d, ex);
}

// Pass 3: agg[dst,:] += alpha * h[src,:].  16 lanes/edge x float4/lane.
__global__ __launch_bounds__(256)
void edge_aggregate(const int* __restrict__ ei, int E, int n_nodes,
                    const float* __restrict__ exb, const float* __restrict__ denom,
                    const float* __restrict__ h, float* __restrict__ agg)
{
  const unsigned t = blockIdx.x * blockDim.x + threadIdx.x;
  const int e  = (int)(t >> 4);
  const int f4 = (int)(t & 15u) * 4;
  if (e >= E + n_nodes) return;
  int s, d; edge_sd(ei, E, e, s, d);
  const float alpha = exb[e] / (denom[d] + 1e-16f);
  const float4 hv = *(const float4*)(h + (size_t)s * 64 + f4);
  float* o = agg + (size_t)d * 64 + f4;
  atomicAdd(o + 0, hv.x * alpha);   // GLOBAL_ATOMIC_ADD_F32, no return
  atomicAdd(o + 1, hv.y * alpha);
  atomicAdd(o + 2, hv.z * alpha);
  atomicAdd(o + 3, hv.w * alpha);
}

// In-place: buf[i] = act(buf[i] + bias[i%64])
template<bool RELU>
__global__ __launch_bounds__(256)
void bias_act(float* __restrict__ buf, const float* __restrict__ bias, int n_nodes)
{
  const unsigned i = blockIdx.x * blockDim.x + threadIdx.x;
  if (i >= (unsigned)n_nodes * 64u) return;
  float v = buf[i] + bias[i & 63u];
  if (RELU) v = fmaxf(v, 0.0f);
  buf[i] = v;
}

// Fused head: logits = h2 @ W2[64x40] + b2, then log_softmax. Wave per node:
// lane owns class c0=lane (all) and c1=lane+32 (lanes 0-7).
__global__ __launch_bounds__(256)
void final_logsoftmax(const float* __restrict__ h2, const float* __restrict__ W2,
                      const float* __restrict__ b2, float* __restrict__ out,
                      int n_nodes)
{
  const int node = (int)((blockIdx.x * blockDim.x + threadIdx.x) >> 5);
  const int lane = threadIdx.x & 31;
  if (node >= n_nodes) return;
  const float* hr = h2 + (size_t)node * 64;
  float acc0 = b2[lane];
  float acc1 = (lane < 8) ? b2[lane + 32] : 0.0f;
#pragma unroll 8
  for (int k = 0; k < 64; ++k) {
    const float hv = hr[k];
    acc0 += hv * W2[k * 40 + lane];
    if (lane < 8) acc1 += hv * W2[k * 40 + lane + 32];
  }
  float m = (lane < 8) ? fmaxf(acc0, acc1) : acc0;
#pragma unroll
  for (int off = 16; off; off >>= 1) m = fmaxf(m, __shfl_xor(m, off, 32));
  float sum = expf(acc0 - m) + ((lane < 8) ? expf(acc1 - m) : 0.0f);
#pragma unroll
  for (int off = 16; off; off >>= 1) sum += __shfl_xor(sum, off, 32);
  const float lse = m + logf(sum);
  out[(size_t)node * 40 + lane] = acc0 - lse;
  if (lane < 8) out[(size_t)node * 40 + lane + 32] = acc1 - lse;
}

// ==========================================================================
extern "C" void kernel_launch(void* const* d_in, const int* in_sizes, int n_in,
                              void* d_out, int out_size, void* d_ws, size_t ws_size,
                              hipStream_t stream)
{
  const float* x     = (const float*)d_in[0];
  const int*   ei    = (const int*)d_in[1];      // [2,E] int32
  const float* W1    = (const float*)d_in[2];
  const float* b1    = (const float*)d_in[3];
  const float* Wc0   = (const float*)d_in[4];
  const float* as0   = (const float*)d_in[5];
  const float* ad0   = (const float*)d_in[6];
  const float* bias0 = (const float*)d_in[7];
  const float* Wc1   = (const float*)d_in[8];
  const float* as1   = (const float*)d_in[9];
  const float* ad1   = (const float*)d_in[10];
  const float* bias1 = (const float*)d_in[11];
  const float* W2    = (const float*)d_in[12];
  const float* b2    = (const float*)d_in[13];
  float*       out   = (float*)d_out;

  const int F_IN = 128, HID = 64;
  const int Nn = in_sizes[0] / F_IN;             // 100000
  const int E  = in_sizes[1] / 2;                // 1600000
  const int Ep = E + Nn;                         // with self loops

  // ---- workspace carve-out ----
  char* ws = (char*)d_ws;
  size_t off = 0;
  auto carve = [&](size_t bytes) -> void* {
    void* p = ws + off;
    off = (off + bytes + 255) & ~(size_t)255;
    return p;
  };
  const size_t hbytes = (size_t)Nn * HID * sizeof(float);
  float*    buf0  = (float*)carve(hbytes);       // h1 -> later hW1
  float*    buf1  = (float*)carve(hbytes);       // hW0 -> later agg1/h2
  float*    buf2  = (float*)carve(hbytes);       // agg0 -> X1
  float*    a_s   = (float*)carve((size_t)Nn * 4);
  float*    a_d   = (float*)carve((size_t)Nn * 4);
  unsigned* emaxu = (unsigned*)carve((size_t)Nn * 4);
  float*    denom = (float*)carve((size_t)Nn * 4);
  float*    exb   = (float*)carve((size_t)Ep * 4);
  if (off > ws_size) return;                     // scratch too small: bail

  const int thr = 256;
  const int gGemm  = (Nn + 31) / 32;             // 3125 (exact)
  const int gWaveN = (Nn * 32 + thr - 1) / thr;  // wave per node
  const int gEdge  = (Ep + thr - 1) / thr;
  const int gEdgeAgg = (int)(((long long)Ep * 16 + thr - 1) / thr);
  const int gElem  = (Nn * HID + thr - 1) / thr;

  // ---- layer 1: h1 = relu(x @ W1 + b1) ----
  gemm_wmma_n64<128, true, true><<<gGemm, thr, 0, stream>>>(x, W1, b1, buf0, Nn);

  // ---- GAT conv 0 ----
  gemm_wmma_n64<64, false, false><<<gGemm, thr, 0, stream>>>(buf0, Wc0, nullptr, buf1, Nn);
  att_scores<<<gWaveN, thr, 0, stream>>>(buf1, as0, ad0, a_s, a_d, Nn);
  hipMemsetAsync(emaxu, 0, (size_t)Nn * 4, stream);  // 0 decodes below any real max
  hipMemsetAsync(denom, 0, (size_t)Nn * 4, stream);
  hipMemsetAsync(buf2,  0, hbytes, stream);
  edge_max<<<gEdge, thr, 0, stream>>>(ei, E, Nn, a_s, a_d, emaxu);
  edge_expsum<<<gEdge, thr, 0, stream>>>(ei, E, Nn, a_s, a_d, emaxu, exb, denom);
  edge_aggregate<<<gEdgeAgg, thr, 0, stream>>>(ei, E, Nn, exb, denom, buf1, buf2);
  bias_act<true><<<gElem, thr, 0, stream>>>(buf2, bias0, Nn);   // X1 = relu(agg+bias0)

  // ---- GAT conv 1 ----
  gemm_wmma_n64<64, false, false><<<gGemm, thr, 0, stream>>>(buf2, Wc1, nullptr, buf0, Nn);
  att_scores<<<gWaveN, thr, 0, stream>>>(buf0, as1, ad1, a_s, a_d, Nn);
  hipMemsetAsync(emaxu, 0, (size_t)Nn * 4, stream);
  hipMemsetAsync(denom, 0, (size_t)Nn * 4, stream);
  hipMemsetAsync(buf1,  0, hbytes, stream);
  edge_max<<<gEdge, thr, 0, stream>>>(ei, E, Nn, a_s, a_d, emaxu);
  edge_expsum<<<gEdge, thr, 0, stream>>>(ei, E, Nn, a_s, a_d, emaxu, exb, denom);
  edge_aggregate<<<gEdgeAgg, thr, 0, stream>>>(ei, E, Nn, exb, denom, buf0, buf1);
  bias_act<false><<<gElem, thr, 0, stream>>>(buf1, bias1, Nn);  // h2 = agg+bias1

  // ---- head: logits + log_softmax ----
  final_logsoftmax<<<gWaveN, thr, 0, stream>>>(buf1, W2, b2, out, Nn);
}